// allkernelsdiff_19026705121508
// MI455X (gfx1250) — compile-verified
//
#include <hip/hip_runtime.h>

#define TILE_H   32
#define LDS_ROWS 34      // TILE_H + 2 halo rows
#define W 128
#define H 128
#define D 128
#define NCH 3

typedef float v2f __attribute__((ext_vector_type(2)));

__global__ __launch_bounds__(256) void conv3d_wrap_kernel(
    const float* __restrict__ x,      // (8,1,128,128,128)
    const float* __restrict__ kern,   // (3,3,3,3) = (o,kd,kh,kw)
    float* __restrict__ out)          // (8,3,128,128,128)
{
  __shared__ float tile[3 * LDS_ROWS * W];   // 52,224 B static LDS

  const int t  = threadIdx.x;
  const int h0 = blockIdx.x * TILE_H;
  const int d0 = blockIdx.y;
  const int n  = blockIdx.z;

  // ---------------- Stage 3 wrapped d-planes into LDS asynchronously --------
  // 3 planes x 34 rows x 128 floats = 3264 x b128 chunks (16B each).
  const int nChunks = 3 * LDS_ROWS * (W / 4);          // 3264
  const float* xn = x + (size_t)n * D * H * W;
  const unsigned lds_base = (unsigned)(unsigned long long)(uintptr_t)&tile[0];

  #pragma unroll 1
  for (int i = 0; i < 13; ++i) {                       // 13*256 = 3328 >= 3264
    int idx = t + (i << 8);
    idx = idx < (nChunks - 1) ? idx : (nChunks - 1);   // clamp: benign dup copy
    const int p   = idx / (LDS_ROWS * (W / 4));        // d-plane 0..2
    const int rem = idx - p * (LDS_ROWS * (W / 4));
    const int r   = rem >> 5;                          // lds row 0..33
    const int cw  = (rem & 31) << 2;                   // w start, step 4 floats
    const int dg  = (d0 - 1 + p) & (D - 1);            // wrap d
    const int hg  = (h0 - 1 + r) & (H - 1);            // wrap h
    const unsigned long long gsrc =
        (unsigned long long)(uintptr_t)(xn + ((size_t)dg * H + hg) * W + cw);
    const unsigned ldst =
        lds_base + (unsigned)(((p * LDS_ROWS + r) * W + cw) * 4);
    // CDNA5 async global->LDS copy (ASYNCcnt-tracked), 16B per lane
    asm volatile("global_load_async_to_lds_b128 %0, %1, off"
                 :: "v"(ldst), "v"(gsrc) : "memory");
  }

#if __has_builtin(__builtin_amdgcn_s_wait_asynccnt)
  __builtin_amdgcn_s_wait_asynccnt(0);
#else
  asm volatile("s_wait_asynccnt 0" ::: "memory");
#endif
  __syncthreads();

  // ---------------- Load the 81 uniform weights (scalar regs) ---------------
  float wk[NCH][3][3][3];
  #pragma unroll
  for (int i = 0; i < 81; ++i) ((float*)wk)[i] = kern[i];

  // ---------------- Compute: 2 w-points x 3 channels per thread step --------
  // Lane l -> w pair (2l, 2l+1): b64 coalesced stores, conflict-free LDS banks.
  const int wp = t & 63;          // 64 w-pairs
  const int w0 = wp << 1;
  const int hs = t >> 6;          // 0..3
  const size_t cstride = (size_t)D * H * W;
  float* obase = out + (size_t)n * NCH * cstride + ((size_t)d0 * H) * W + w0;

  #pragma unroll 1
  for (int j = hs; j < TILE_H; j += 4) {
    float a00 = 0.f, a01 = 0.f;   // ch0 @ (w0, w0+1)
    float a10 = 0.f, a11 = 0.f;   // ch1
    float a20 = 0.f, a21 = 0.f;   // ch2
    #pragma unroll
    for (int dd = 0; dd < 3; ++dd) {
      #pragma unroll
      for (int dh = 0; dh < 3; ++dh) {
        const float* row = &tile[(dd * LDS_ROWS + j + dh) * W];
        const float m0 = row[(w0 + (W - 1)) & (W - 1)];   // w0-1 (wrap)
        const float m1 = row[w0];
        const float m2 = row[w0 + 1];
        const float m3 = row[(w0 + 2) & (W - 1)];         // w0+2 (wrap)
        #pragma unroll
        for (int dw = 0; dw < 3; ++dw) {
          const float vL = (dw == 0) ? m0 : (dw == 1) ? m1 : m2;
          const float vR = (dw == 0) ? m1 : (dw == 1) ? m2 : m3;
          a00 = fmaf(wk[0][dd][dh][dw], vL, a00);
          a01 = fmaf(wk[0][dd][dh][dw], vR, a01);
          a10 = fmaf(wk[1][dd][dh][dw], vL, a10);
          a11 = fmaf(wk[1][dd][dh][dw], vR, a11);
          a20 = fmaf(wk[2][dd][dh][dw], vL, a20);
          a21 = fmaf(wk[2][dd][dh][dw], vR, a21);
        }
      }
    }
    float* o = obase + (size_t)(h0 + j) * W;
    const v2f p0 = {a00, a01}, p1 = {a10, a11}, p2 = {a20, a21};
    __builtin_nontemporal_store(p0, (v2f*)o);             // streamed, 8B aligned
    __builtin_nontemporal_store(p1, (v2f*)(o + cstride));
    __builtin_nontemporal_store(p2, (v2f*)(o + 2 * cstride));
  }
}

extern "C" void kernel_launch(void* const* d_in, const int* in_sizes, int n_in,
                              void* d_out, int out_size, void* d_ws, size_t ws_size,
                              hipStream_t stream) {
  const float* x    = (const float*)d_in[0];   // 8*128^3
  const float* kern = (const float*)d_in[1];   // 81
  float* out        = (float*)d_out;           // 8*3*128^3
  (void)in_sizes; (void)n_in; (void)out_size; (void)d_ws; (void)ws_size;

  dim3 grid(H / TILE_H, D, 8);                 // 4 x 128 x 8 = 4096 blocks
  conv3d_wrap_kernel<<<grid, 256, 0, stream>>>(x, kern, out);
}